// NTM_78726750535824
// MI455X (gfx1250) — compile-verified
//
#include <hip/hip_runtime.h>
#include <hip/hip_bf16.h>
#include <math.h>

// ---------------- problem constants ----------------
static constexpr int kB   = 64;
static constexpr int kN   = 8192;
static constexpr int kM   = 128;
static constexpr int kH   = 512;
static constexpr int k4H  = 2048;
static constexpr int kNumIn = 256;
static constexpr int kInDim = 384;   // M + NUM_IN
static constexpr int kP   = 390;     // 3 + 3*M + SHIFT

// d_out offsets (floats), concatenated in reference return order
static constexpr size_t OUT_OFF = 0;                       // 64*256
static constexpr size_t R_OFF   = 16384;                   // 64*128
static constexpr size_t MEM_OFF = 24576;                   // 64*8192*128
static constexpr size_t W_OFF   = MEM_OFF + (size_t)kB * kN; // placeholder fixed below
static constexpr size_t MEMSZ   = (size_t)kB * kN * kM;    // 67108864
static constexpr size_t WGT_OFF = MEM_OFF + MEMSZ;         // 67133440

// workspace layout (floats)
static constexpr size_t WS_GATES = 0;                        // 64*2048
static constexpr size_t WS_H     = WS_GATES + (size_t)kB*k4H;     // 32768
static constexpr size_t WS_P     = WS_H + (size_t)kB*kH;          // 64*390
static constexpr size_t WS_KE    = WS_P + (size_t)kB*kP;          // 64*128
static constexpr size_t WS_E     = WS_KE + (size_t)kB*kM;
static constexpr size_t WS_A     = WS_E + (size_t)kB*kM;
static constexpr size_t WS_BETA  = WS_A + (size_t)kB*kM;
static constexpr size_t WS_GAMMA = WS_BETA + kB;
static constexpr size_t WS_G     = WS_GAMMA + kB;
static constexpr size_t WS_S0    = WS_G + kB;
static constexpr size_t WS_S1    = WS_S0 + kB;
static constexpr size_t WS_S2    = WS_S1 + kB;
static constexpr size_t WS_KNORM = WS_S2 + kB;
static constexpr size_t WS_SCORES= WS_KNORM + kB;                 // 64*8192
static constexpr size_t WS_RPART = WS_SCORES + (size_t)kB*kN;     // 64*32*128

typedef __attribute__((ext_vector_type(2))) float v2f;
typedef __attribute__((ext_vector_type(8))) float v8f;

__device__ __forceinline__ float sigm(float x)     { return 1.0f / (1.0f + __expf(-x)); }
__device__ __forceinline__ float softplus(float x) { return (x > 20.0f) ? x : log1pf(__expf(x)); }

// =====================================================================
// 1) gates = concat(r,x) @ W_ih^T + h @ W_hh^T + b_ih + b_hh
//    f32 WMMA 16x16x4. 512 tiles (4 row-tiles x 128 col-tiles), 1 wave/tile.
// =====================================================================
__global__ __launch_bounds__(256) void gates_kernel(
    const float* __restrict__ r, const float* __restrict__ x,
    const float* __restrict__ h,
    const float* __restrict__ Wih, const float* __restrict__ Whh,
    const float* __restrict__ bih, const float* __restrict__ bhh,
    float* __restrict__ gates)
{
    const int wave = blockIdx.x * 8 + (threadIdx.x >> 5);   // 0..511
    const int lane = threadIdx.x & 31;
    const int tm = wave & 3;          // row tile (0..3)
    const int tn = wave >> 2;         // col tile (0..127)
    const int half = lane >> 4;       // k-subgroup (0/1)
    const int l16 = lane & 15;
    const int arow = tm * 16 + l16;   // 0..63
    const int bcol = tn * 16 + l16;   // 0..2047

    v8f acc = {};
    // ---- A = concat(r, x) over K = 384 ----
    for (int k = 0; k < kInDim; k += 4) {
        const int ka = k + half * 2;
        v2f a;
        if (ka < kM) a = *(const v2f*)(r + arow * kM + ka);
        else         a = *(const v2f*)(x + arow * kNumIn + (ka - kM));
        v2f b = *(const v2f*)(Wih + bcol * kInDim + ka);
        acc = __builtin_amdgcn_wmma_f32_16x16x4_f32(false, a, false, b,
                                                    (short)0, acc, false, false);
    }
    // ---- A = lstm_h over K = 512 ----
    for (int k = 0; k < kH; k += 4) {
        const int ka = k + half * 2;
        v2f a = *(const v2f*)(h + arow * kH + ka);
        v2f b = *(const v2f*)(Whh + bcol * kH + ka);
        acc = __builtin_amdgcn_wmma_f32_16x16x4_f32(false, a, false, b,
                                                    (short)0, acc, false, false);
    }
    const float bias = bih[bcol] + bhh[bcol];
    #pragma unroll
    for (int v = 0; v < 8; ++v) {
        const int row = tm * 16 + v + half * 8;          // C/D VGPR layout
        gates[row * k4H + bcol] = acc[v] + bias;
    }
}

// =====================================================================
// 2) elementwise LSTM -> h_new
// =====================================================================
__global__ void lstm_kernel(const float* __restrict__ gates,
                            const float* __restrict__ c_old,
                            float* __restrict__ h_new)
{
    const int i = blockIdx.x * blockDim.x + threadIdx.x;
    if (i >= kB * kH) return;
    const int b = i >> 9;
    const int hc = i & (kH - 1);
    const float* gr = gates + b * k4H;
    const float ig = sigm(gr[hc]);
    const float fg = sigm(gr[kH + hc]);
    const float gg = tanhf(gr[2 * kH + hc]);
    const float og = sigm(gr[3 * kH + hc]);
    const float c = fg * c_old[i] + ig * gg;
    h_new[i] = og * tanhf(c);
}

// =====================================================================
// 3/4) generic C = act(h_new @ W^T + bias), K = 512, WMMA f32 16x16x4
//      ACT: 0 = identity (p head), 1 = sigmoid (output head)
// =====================================================================
template <int ACT>
__global__ __launch_bounds__(256) void gemm_hx(
    const float* __restrict__ A,      // 64 x 512
    const float* __restrict__ W,      // Ncols x 512
    const float* __restrict__ bias,   // Ncols
    float* __restrict__ C,            // 64 x ldc
    int Ncols, int NtilesN, int ldc)
{
    const int wave = blockIdx.x * 8 + (threadIdx.x >> 5);
    if (wave >= 4 * NtilesN) return;               // whole-wave uniform exit
    const int lane = threadIdx.x & 31;
    const int tm = wave & 3;
    const int tn = wave >> 2;
    const int half = lane >> 4;
    const int l16 = lane & 15;
    const int arow = tm * 16 + l16;
    const int bcol = tn * 16 + l16;
    const bool bok = (bcol < Ncols);

    v8f acc = {};
    for (int k = 0; k < kH; k += 4) {
        const int ka = k + half * 2;
        v2f a = *(const v2f*)(A + arow * kH + ka);
        v2f b;
        if (bok) b = *(const v2f*)(W + bcol * kH + ka);
        else     { b[0] = 0.0f; b[1] = 0.0f; }
        acc = __builtin_amdgcn_wmma_f32_16x16x4_f32(false, a, false, b,
                                                    (short)0, acc, false, false);
    }
    if (!bok) return;
    const float bs = bias[bcol];
    #pragma unroll
    for (int v = 0; v < 8; ++v) {
        const int row = tm * 16 + v + half * 8;
        float val = acc[v] + bs;
        if (ACT == 1) val = sigm(val);
        C[row * ldc + bcol] = val;
    }
}

// =====================================================================
// 5) parse p -> beta, gamma, g, softmax(s), kappa+eps (+norm), e, a
// =====================================================================
__global__ __launch_bounds__(128) void prep_kernel(
    const float* __restrict__ p,
    float* __restrict__ ke, float* __restrict__ ev, float* __restrict__ av,
    float* __restrict__ beta, float* __restrict__ gamma, float* __restrict__ g,
    float* __restrict__ s0, float* __restrict__ s1, float* __restrict__ s2,
    float* __restrict__ knorm)
{
    __shared__ float red[128];
    const int b = blockIdx.x, t = threadIdx.x;
    const float* pr = p + b * kP;
    const float kv = pr[1 + t] + 1e-12f;           // kappa slice [1, 129)
    ke[b * kM + t] = kv;
    ev[b * kM + t] = sigm(pr[134 + t]);            // e slice [134, 262)
    av[b * kM + t] = sigm(pr[262 + t]);            // a slice [262, 390)
    red[t] = kv * kv;
    __syncthreads();
    for (int s = 64; s > 0; s >>= 1) { if (t < s) red[t] += red[t + s]; __syncthreads(); }
    if (t == 0) {
        knorm[b] = sqrtf(red[0]);
        beta[b]  = softplus(pr[0]);
        gamma[b] = 1.0f + softplus(pr[129]);
        g[b]     = sigm(pr[130]);
        const float q0 = softplus(pr[131]), q1 = softplus(pr[132]), q2 = softplus(pr[133]);
        const float mx = fmaxf(q0, fmaxf(q1, q2));
        const float e0 = __expf(q0 - mx), e1 = __expf(q1 - mx), e2 = __expf(q2 - mx);
        const float inv = 1.0f / (e0 + e1 + e2);
        s0[b] = e0 * inv; s1[b] = e1 * inv; s2[b] = e2 * inv;
    }
}

// =====================================================================
// 6) cosine-similarity scores: beta * <mem_e, k_e> / max(|mem_e||k_e|,1e-8)
//    one wave per (b,n) row; float4 loads; wave32 shuffle reduce
// =====================================================================
__global__ __launch_bounds__(256) void score_kernel(
    const float* __restrict__ mem, const float* __restrict__ ke,
    const float* __restrict__ beta, const float* __restrict__ knorm,
    float* __restrict__ scores)
{
    const int b = blockIdx.y;
    __shared__ float kesh[kM];
    __shared__ float sb[2];
    if (threadIdx.x < kM) kesh[threadIdx.x] = ke[b * kM + threadIdx.x];
    if (threadIdx.x == 0) { sb[0] = beta[b]; sb[1] = knorm[b]; }
    __syncthreads();
    const int wave = threadIdx.x >> 5, lane = threadIdx.x & 31;
    const float4 kef = *(const float4*)(kesh + lane * 4);
    #pragma unroll 1
    for (int i = 0; i < 8; ++i) {
        const int n = blockIdx.x * 64 + i * 8 + wave;
        const float* mrow = mem + ((size_t)b * kN + n) * kM;
        __builtin_prefetch(mrow + 8 * kM + lane * 4, 0, 0);   // next row for this wave
        const float4 mv = *(const float4*)(mrow + lane * 4);
        const float m0 = mv.x + 1e-12f, m1 = mv.y + 1e-12f,
                    m2 = mv.z + 1e-12f, m3 = mv.w + 1e-12f;
        float dot = m0 * kef.x + m1 * kef.y + m2 * kef.z + m3 * kef.w;
        float nsq = m0 * m0 + m1 * m1 + m2 * m2 + m3 * m3;
        #pragma unroll
        for (int off = 16; off > 0; off >>= 1) {
            dot += __shfl_xor(dot, off, 32);
            nsq += __shfl_xor(nsq, off, 32);
        }
        if (lane == 0) {
            const float denom = fmaxf(sqrtf(nsq) * sb[1], 1e-8f);
            scores[b * kN + n] = sb[0] * dot / denom;
        }
    }
}

// =====================================================================
// 7) per-b row pipeline: softmax -> interpolate -> circular shift -> sharpen
//    one 512-thread WG per batch row; 8192-float row lives in LDS
// =====================================================================
__global__ __launch_bounds__(512) void weight_kernel(
    const float* __restrict__ scores, const float* __restrict__ w_prev,
    const float* __restrict__ gp, const float* __restrict__ gammap,
    const float* __restrict__ s0p, const float* __restrict__ s1p,
    const float* __restrict__ s2p, float* __restrict__ weight)
{
    __shared__ float sc[kN];
    __shared__ float red[512];
    const int b = blockIdx.x, t = threadIdx.x;
    const float g = gp[b], gamma = gammap[b];
    const float s0 = s0p[b], s1 = s1p[b], s2 = s2p[b];

    float lmax = -INFINITY;
    #pragma unroll
    for (int i = 0; i < 16; ++i) {
        const int n = t + i * 512;
        const float v = scores[b * kN + n];
        sc[n] = v;
        lmax = fmaxf(lmax, v);
    }
    red[t] = lmax; __syncthreads();
    for (int s = 256; s > 0; s >>= 1) { if (t < s) red[t] = fmaxf(red[t], red[t + s]); __syncthreads(); }
    const float rowmax = red[0]; __syncthreads();

    float lsum = 0.0f;
    #pragma unroll
    for (int i = 0; i < 16; ++i) {
        const int n = t + i * 512;
        const float e = __expf(sc[n] - rowmax);
        sc[n] = e; lsum += e;
    }
    red[t] = lsum; __syncthreads();
    for (int s = 256; s > 0; s >>= 1) { if (t < s) red[t] += red[t + s]; __syncthreads(); }
    const float invtot = 1.0f / red[0]; __syncthreads();

    #pragma unroll
    for (int i = 0; i < 16; ++i) {
        const int n = t + i * 512;
        const float wc = sc[n] * invtot;
        sc[n] = g * wc + (1.0f - g) * w_prev[b * kN + n];
    }
    __syncthreads();

    // w_hat[n] = s0*wg[n-1] + s1*wg[n] + s2*wg[n+1] (circular), then ^gamma
    float wpow[16];
    float lps = 0.0f;
    const int base = t * 16;
    #pragma unroll
    for (int i = 0; i < 16; ++i) {
        const int n = base + i;
        const float wh = s0 * sc[(n + kN - 1) & (kN - 1)]
                       + s1 * sc[n]
                       + s2 * sc[(n + 1) & (kN - 1)];
        const float pw = __powf(wh, gamma);
        wpow[i] = pw; lps += pw;
    }
    red[t] = lps; __syncthreads();
    for (int s = 256; s > 0; s >>= 1) { if (t < s) red[t] += red[t + s]; __syncthreads(); }
    const float winv = 1.0f / (red[0] + 1e-12f);
    #pragma unroll
    for (int i = 0; i < 16; i += 4) {
        float4 o;
        o.x = wpow[i + 0] * winv; o.y = wpow[i + 1] * winv;
        o.z = wpow[i + 2] * winv; o.w = wpow[i + 3] * winv;
        *(float4*)(weight + (size_t)b * kN + base + i) = o;
    }
}

// =====================================================================
// 8) memory update + deterministic partial r_new reduction
//    mem_new = mem*(1 - w*e) + w*a ;  rpart[b,chunk,:] = sum_n w*mem_new
// =====================================================================
__global__ __launch_bounds__(256) void update_kernel(
    const float* __restrict__ mem, const float* __restrict__ weight,
    const float* __restrict__ ev, const float* __restrict__ av,
    float* __restrict__ mem_new, float* __restrict__ rpart)
{
    const int b = blockIdx.y;
    const int chunk = blockIdx.x;          // 32 chunks of 256 rows
    __shared__ float esh[kM], ash[kM];
    __shared__ float racc[8 * kM];
    const int t = threadIdx.x;
    if (t < kM) { esh[t] = ev[b * kM + t]; ash[t] = av[b * kM + t]; }
    __syncthreads();
    const int wave = t >> 5, lane = t & 31;
    const float4 ef = *(const float4*)(esh + lane * 4);
    const float4 af = *(const float4*)(ash + lane * 4);
    float4 rs = {0.0f, 0.0f, 0.0f, 0.0f};
    #pragma unroll 1
    for (int i = 0; i < 32; ++i) {
        const int n = chunk * 256 + i * 8 + wave;
        const size_t rb = ((size_t)b * kN + n) * kM;
        const float w = weight[(size_t)b * kN + n];
        const float* mrow = mem + rb;
        __builtin_prefetch(mrow + 8 * kM + lane * 4, 0, 0);
        const float4 mv = *(const float4*)(mrow + lane * 4);
        float4 mn;
        mn.x = mv.x * (1.0f - w * ef.x) + w * af.x;
        mn.y = mv.y * (1.0f - w * ef.y) + w * af.y;
        mn.z = mv.z * (1.0f - w * ef.z) + w * af.z;
        mn.w = mv.w * (1.0f - w * ef.w) + w * af.w;
        *(float4*)(mem_new + rb + lane * 4) = mn;
        rs.x += w * mn.x; rs.y += w * mn.y; rs.z += w * mn.z; rs.w += w * mn.w;
    }
    *(float4*)(racc + wave * kM + lane * 4) = rs;
    __syncthreads();
    if (t < kM) {
        float s = 0.0f;
        #pragma unroll
        for (int wv = 0; wv < 8; ++wv) s += racc[wv * kM + t];
        rpart[((size_t)b * 32 + chunk) * kM + t] = s;
    }
}

// 9) final r_new reduce (deterministic)
__global__ void rreduce_kernel(const float* __restrict__ rpart, float* __restrict__ rout)
{
    const int b = blockIdx.x, m = threadIdx.x;
    float s = 0.0f;
    #pragma unroll
    for (int c = 0; c < 32; ++c) s += rpart[((size_t)b * 32 + c) * kM + m];
    rout[b * kM + m] = s;
}

// =====================================================================
extern "C" void kernel_launch(void* const* d_in, const int* in_sizes, int n_in,
                              void* d_out, int out_size, void* d_ws, size_t ws_size,
                              hipStream_t stream) {
    (void)in_sizes; (void)n_in; (void)out_size; (void)ws_size;
    const float* x        = (const float*)d_in[0];
    const float* r        = (const float*)d_in[1];
    const float* lstm_h   = (const float*)d_in[2];
    const float* lstm_c   = (const float*)d_in[3];
    const float* memory   = (const float*)d_in[4];
    const float* w_prev   = (const float*)d_in[5];
    const float* W_ih     = (const float*)d_in[6];
    const float* W_hh     = (const float*)d_in[7];
    const float* b_ih     = (const float*)d_in[8];
    const float* b_hh     = (const float*)d_in[9];
    const float* W_params = (const float*)d_in[10];
    const float* b_params = (const float*)d_in[11];
    const float* W_fc     = (const float*)d_in[12];
    const float* b_fc     = (const float*)d_in[13];

    float* out = (float*)d_out;
    float* ws  = (float*)d_ws;

    float* g_gates  = ws + WS_GATES;
    float* g_h      = ws + WS_H;
    float* g_p      = ws + WS_P;
    float* g_ke     = ws + WS_KE;
    float* g_e      = ws + WS_E;
    float* g_a      = ws + WS_A;
    float* g_beta   = ws + WS_BETA;
    float* g_gamma  = ws + WS_GAMMA;
    float* g_g      = ws + WS_G;
    float* g_s0     = ws + WS_S0;
    float* g_s1     = ws + WS_S1;
    float* g_s2     = ws + WS_S2;
    float* g_knorm  = ws + WS_KNORM;
    float* g_scores = ws + WS_SCORES;
    float* g_rpart  = ws + WS_RPART;

    float* o_out    = out + OUT_OFF;
    float* o_r      = out + R_OFF;
    float* o_mem    = out + MEM_OFF;
    float* o_weight = out + WGT_OFF;

    // 1) gates GEMM (WMMA f32): 512 tiles, 8 waves/block -> 64 blocks
    gates_kernel<<<64, 256, 0, stream>>>(r, x, lstm_h, W_ih, W_hh, b_ih, b_hh, g_gates);
    // 2) LSTM elementwise -> h_new
    lstm_kernel<<<(kB * kH + 255) / 256, 256, 0, stream>>>(g_gates, lstm_c, g_h);
    // 3) p head: N=390 -> 25 col tiles -> 100 waves -> 13 blocks
    gemm_hx<0><<<13, 256, 0, stream>>>(g_h, W_params, b_params, g_p, kP, 25, kP);
    // 4) output head: N=256 -> 16 col tiles -> 64 waves -> 8 blocks
    gemm_hx<1><<<8, 256, 0, stream>>>(g_h, W_fc, b_fc, o_out, 256, 16, 256);
    // 5) parse p
    prep_kernel<<<kB, 128, 0, stream>>>(g_p, g_ke, g_e, g_a, g_beta, g_gamma, g_g,
                                        g_s0, g_s1, g_s2, g_knorm);
    // 6) content addressing scores (streams 256 MB)
    score_kernel<<<dim3(128, kB), 256, 0, stream>>>(memory, g_ke, g_beta, g_knorm, g_scores);
    // 7) softmax + interpolate + shift + sharpen -> weight
    weight_kernel<<<kB, 512, 0, stream>>>(g_scores, w_prev, g_g, g_gamma,
                                          g_s0, g_s1, g_s2, o_weight);
    // 8) memory update + partial r (streams 512 MB)
    update_kernel<<<dim3(32, kB), 256, 0, stream>>>(memory, o_weight, g_e, g_a,
                                                    o_mem, g_rpart);
    // 9) r_new
    rreduce_kernel<<<kB, 128, 0, stream>>>(g_rpart, o_r);
}